// CaT_13941463842986
// MI455X (gfx1250) — compile-verified
//
#include <hip/hip_runtime.h>
#include <cstdint>

// Problem constants from the reference: B=512, T=128, H=8, L=3.
#define TT 128
#define HH 8
#define LL 3
#define NEGID (-1e30f)   // matches reference NEG stand-in for -inf

// Online-softmax scan state: running max m, running sum Z, running weighted sum S.
struct SMS { float m, z, s; };

__device__ __forceinline__ SMS sms_ident() {
  SMS r; r.m = NEGID; r.z = 0.f; r.s = 0.f; return r;
}

// Associative, commutative combine for the softmax prefix scan.
__device__ __forceinline__ SMS sms_combine(const SMS a, const SMS b) {
  SMS r;
  r.m = fmaxf(a.m, b.m);
  float ea = __expf(a.m - r.m);   // args always <= 0 -> no overflow
  float eb = __expf(b.m - r.m);
  r.z = fmaf(a.z, ea, b.z * eb);
  r.s = fmaf(a.s, ea, b.s * eb);
  return r;
}

__device__ __forceinline__ SMS sms_shfl_up(const SMS a, int d) {
  SMS r;
  r.m = __shfl_up(a.m, d, 32);
  r.z = __shfl_up(a.z, d, 32);
  r.s = __shfl_up(a.s, d, 32);
  return r;
}

__global__ __launch_bounds__(256) void cat_fused_kernel(
    const float* __restrict__ X,
    const float* __restrict__ wk, const float* __restrict__ wq,
    const float* __restrict__ wv,
    const float* __restrict__ Wp, const float* __restrict__ bp,
    const float* __restrict__ W1, const float* __restrict__ b1,
    const float* __restrict__ W2, const float* __restrict__ b2,
    const float* __restrict__ w_lm, const float* __restrict__ b_lm,
    float* __restrict__ out)
{
  __shared__ __align__(16) float sx[TT];  // current residual stream x[b,:]
  __shared__ float sy[TT];                // per-layer y accumulator

  const int tid  = threadIdx.x;
  const int lane = tid & 31;
  const int wid  = tid >> 5;              // wave id == head id (8 waves)
  const int b    = blockIdx.x;

  // ---- Stage X[b, 0:128] into LDS via the CDNA5 async global->LDS path. ----
  // Wave 0: each of 32 lanes issues one b128 async copy (4 floats) -> 512 B row.
  if (wid == 0) {
    uint64_t gaddr = (uint64_t)(uintptr_t)(X + (size_t)b * TT + lane * 4);
    // Low 32 bits of the flat shared address are the wave-relative LDS offset.
    uint32_t laddr = (uint32_t)(uintptr_t)(&sx[lane * 4]);
    asm volatile("global_load_async_to_lds_b128 %0, %1, off"
                 :: "v"(laddr), "v"(gaddr) : "memory");
    asm volatile("s_wait_asynccnt 0" ::: "memory");
  }
  __syncthreads();

  const float wlm = w_lm[0];
  const float blm = b_lm[0];

  for (int l = 0; l < LL; ++l) {
    // Init y accumulator with the projection bias.
    if (tid < TT) sy[tid] = bp[l];

    // Per-head scalar weights (uniform within each wave).
    const int   wi  = l * HH + wid;
    const float aqk = wk[wi] * wq[wi];   // qk_j = x_j^2 * aqk
    const float wvh = wv[wi];            // v_j  = x_j * wvh
    const float wph = Wp[wi];            // output projection weight for head

    __syncthreads();

    // Each lane owns 4 consecutive t values: t = lane*4 + k.  ds_load_b128.
    float4 xv = *(const float4*)&sx[lane * 4];
    float xs[4] = {xv.x, xv.y, xv.z, xv.w};

    // In-lane sequential exclusive scan (depth 4), keeping exclusive states.
    SMS ex[4];
    SMS st = sms_ident();
#pragma unroll
    for (int k = 0; k < 4; ++k) {
      ex[k] = st;
      float q  = xs[k] * xs[k] * aqk;
      float vv = xs[k] * wvh;
      float nm = fmaxf(st.m, q);
      float es = __expf(st.m - nm);
      float eq = __expf(q - nm);
      st.m = nm;
      st.z = fmaf(st.z, es, eq);
      st.s = fmaf(st.s, es, vv * eq);
    }

    // Hillis-Steele inclusive scan of lane totals across the wave32 (depth 5).
    SMS inc = st;
#pragma unroll
    for (int d = 1; d < 32; d <<= 1) {
      SMS p = sms_shfl_up(inc, d);
      if (lane >= d) inc = sms_combine(p, inc);
    }
    // Exclusive lane prefix.
    SMS exl = sms_shfl_up(inc, 1);
    if (lane == 0) exl = sms_ident();

    // Emit head_out[t] = S/Z over j < t; row t=0 has no parents -> 0.
#pragma unroll
    for (int k = 0; k < 4; ++k) {
      SMS pre = sms_combine(exl, ex[k]);
      int   t  = lane * 4 + k;
      float ho = (t == 0) ? 0.f : pre.s / fmaxf(pre.z, 1e-30f);
      atomicAdd(&sy[t], ho * wph);   // ds_add_f32, 8 heads per element
    }
    __syncthreads();

    // Fused output projection done; now FF (1->4->1) + residual, back into sx.
    if (tid < TT) {
      float y = sy[tid];
      float f = b2[l];
#pragma unroll
      for (int u = 0; u < 4; ++u) {
        float h1 = fmaxf(fmaf(y, W1[l * 4 + u], b1[l * 4 + u]), 0.f);
        f = fmaf(h1, W2[l * 4 + u], f);
      }
      sx[tid] = y + f;
    }
    __syncthreads();
  }

  // lm_head: out = x * w_lm + b_lm
  if (tid < TT) out[(size_t)b * TT + tid] = fmaf(sx[tid], wlm, blm);
}

extern "C" void kernel_launch(void* const* d_in, const int* in_sizes, int n_in,
                              void* d_out, int out_size, void* d_ws, size_t ws_size,
                              hipStream_t stream) {
  (void)n_in; (void)d_ws; (void)ws_size; (void)out_size;
  const float* X    = (const float*)d_in[0];
  const float* wk   = (const float*)d_in[1];
  const float* wq   = (const float*)d_in[2];
  const float* wv   = (const float*)d_in[3];
  const float* Wp   = (const float*)d_in[4];
  const float* bp   = (const float*)d_in[5];
  const float* W1   = (const float*)d_in[6];
  const float* b1   = (const float*)d_in[7];
  const float* W2   = (const float*)d_in[8];
  const float* b2   = (const float*)d_in[9];
  const float* w_lm = (const float*)d_in[10];
  const float* b_lm = (const float*)d_in[11];

  const int B = in_sizes[0] / TT;  // 512
  cat_fused_kernel<<<B, 256, 0, stream>>>(
      X, wk, wq, wv, Wp, bp, W1, b1, W2, b2, w_lm, b_lm, (float*)d_out);
}